// DilatedAttention_79044578116355
// MI455X (gfx1250) — compile-verified
//
#include <hip/hip_runtime.h>

// Dilated 3x3 neighborhood attention, B=8, DIM=256, H=W=64, dilation=2, pad=2.
// One workgroup (128 threads = 4 wave32) handles one (batch, row, 16-pixel strip).
// QK scores via V_WMMA_F32_16X16X4_F32; staging via GLOBAL_LOAD_ASYNC_TO_LDS_B32
// (CDNA5 async memory->LDS path, tracked with ASYNCcnt); softmax + AV on VALU.

#define BB    8
#define DIM   256
#define HH    64
#define WW    64
#define NN    (HH * WW)
#define TILE  16      // pixels per strip
#define SPAN  20      // staged key/value columns: x0-2 .. x0+17
#define KR    3       // key rows (y-2, y, y+2)

typedef __attribute__((ext_vector_type(2))) float v2f;
typedef __attribute__((ext_vector_type(8))) float v8f;

#if __has_builtin(__builtin_amdgcn_global_load_async_to_lds_b32) && \
    __has_builtin(__builtin_amdgcn_s_wait_asynccnt)
#define USE_ASYNC_LDS 1
#else
#define USE_ASYNC_LDS 0
#endif

#if USE_ASYNC_LDS
#define GLD_ASYNC_B32(gp, lp)                                              \
    __builtin_amdgcn_global_load_async_to_lds_b32(                         \
        (__attribute__((address_space(1))) int*)(gp),                      \
        (__attribute__((address_space(3))) int*)(lp), 0, 0)
#endif

__global__ __launch_bounds__(128)
void dilated_attn_kernel(const float* __restrict__ qg,
                         const float* __restrict__ kg,
                         const float* __restrict__ vg,
                         float* __restrict__ og) {
    __shared__ float Qs[DIM * TILE];          // [chan][pixel]          16 KB
    __shared__ float Ks[KR * DIM * SPAN];     // [row][chan][spancol]   60 KB
    __shared__ float Vs[KR * DIM * SPAN];     //                        60 KB
    __shared__ float Sbuf[6 * 16 * 16];       // 6 score tiles           6 KB
    __shared__ float Attn[TILE * 9];

    const int t    = threadIdx.x;
    const int lane = t & 31;
    const int wave = t >> 5;

    const int bid = blockIdx.x;
    const int x0  = (bid & 3) * TILE;         // W/TILE = 4 strips
    const int y   = (bid >> 2) & (HH - 1);
    const int b   = bid >> 8;

    // ---- zero score accumulation buffer ----
    for (int i = t; i < 6 * 16 * 16; i += 128) Sbuf[i] = 0.0f;

    // ---- stage Q tile: Qs[d*16 + p] = q[b, d, y, x0+p] (always in-bounds) ----
    {
        const float* qb = qg + (size_t)b * DIM * NN + (size_t)y * WW + x0;
        for (int idx = t; idx < DIM * TILE; idx += 128) {
            int d = idx >> 4, p = idx & 15;
#if USE_ASYNC_LDS
            GLD_ASYNC_B32(qb + (size_t)d * NN + p, &Qs[idx]);
#else
            Qs[idx] = qb[(size_t)d * NN + p];
#endif
        }
    }

    // ---- stage K/V spans; padded taps must be exactly 0 (reference zero-pads) ----
    for (int idx = t; idx < KR * DIM * SPAN; idx += 128) {
        int scol = idx % SPAN;
        int rest = idx / SPAN;
        int d = rest & (DIM - 1);
        int i = rest >> 8;                    // key row index 0..2
        int yk = y + 2 * (i - 1);
        int xk = x0 - 2 + scol;
        bool inb = ((unsigned)yk < (unsigned)HH) && ((unsigned)xk < (unsigned)WW);
#if USE_ASYNC_LDS
        if (inb) {
            size_t g = ((size_t)b * DIM + d) * NN + (size_t)yk * WW + xk;
            GLD_ASYNC_B32(kg + g, &Ks[idx]);
            GLD_ASYNC_B32(vg + g, &Vs[idx]);
        } else {
            Ks[idx] = 0.0f;
            Vs[idx] = 0.0f;
        }
#else
        float kv = 0.0f, vv = 0.0f;
        if (inb) {
            size_t g = ((size_t)b * DIM + d) * NN + (size_t)yk * WW + xk;
            kv = kg[g];
            vv = vg[g];
        }
        Ks[idx] = kv;
        Vs[idx] = vv;
#endif
    }
#if USE_ASYNC_LDS
    __builtin_amdgcn_s_wait_asynccnt(0);      // ASYNCcnt not covered by barrier
#endif
    __syncthreads();

    // ---- QK scores via WMMA f32 16x16x4, contraction over channels ----
    // tile t6 = keyrow*2 + half; half 0 -> span cols 0..15, half 1 -> cols 4..19
    {
        v8f acc[6] = {};
        const int p    = lane & 15;            // M for A, N for B
        const int koff = (lane >> 4) << 1;     // hi half of wave handles K+2,K+3
        const int c0   = wave * (DIM / 4);     // 64 channels per wave

        for (int c = c0; c < c0 + DIM / 4; c += 4) {
            v2f a;
            a.x = Qs[(c + koff) * TILE + p];         // A[M=p, K]   (VGPR0: K0|K2)
            a.y = Qs[(c + koff + 1) * TILE + p];     //             (VGPR1: K1|K3)
            #pragma unroll
            for (int t6 = 0; t6 < 6; ++t6) {
                const int i   = t6 >> 1;
                const int col = (t6 & 1) * 4 + p;    // span column for lane N
                const float* kb = &Ks[(i * DIM + c + koff) * SPAN + col];
                v2f bm;
                bm.x = kb[0];                         // B[K, N=col]
                bm.y = kb[SPAN];
                acc[t6] = __builtin_amdgcn_wmma_f32_16x16x4_f32(
                    false, a, false, bm, (short)0, acc[t6], false, false);
            }
        }

        // D layout: VGPR r -> M = r (lanes 0-15) / r+8 (lanes 16-31), N = lane%16
        const int mo = (lane >> 4) << 3;
        #pragma unroll
        for (int t6 = 0; t6 < 6; ++t6)
            #pragma unroll
            for (int r = 0; r < 8; ++r)
                atomicAdd(&Sbuf[(t6 * 16 + r + mo) * 16 + (lane & 15)], acc[t6][r]);
    }
    __syncthreads();

    // ---- softmax over the 9 taps (threads 0..15, one pixel each) ----
    if (t < TILE) {
        const int p = t;
        float sc[9];
        #pragma unroll
        for (int i = 0; i < 3; ++i)
            #pragma unroll
            for (int j = 0; j < 3; ++j) {
                int s = p + 2 * j;                 // span col = p + 2j
                int h = (s < 16) ? 0 : 1;
                int col = s - h * 4;               // tile-local N
                sc[i * 3 + j] = Sbuf[((i * 2 + h) * 16 + p) * 16 + col] * 0.0625f;
            }
        float mx = sc[0];
        #pragma unroll
        for (int k = 1; k < 9; ++k) mx = fmaxf(mx, sc[k]);
        float sum = 0.0f;
        #pragma unroll
        for (int k = 0; k < 9; ++k) { sc[k] = __expf(sc[k] - mx); sum += sc[k]; }
        float inv = 1.0f / sum;
        #pragma unroll
        for (int k = 0; k < 9; ++k) Attn[p * 9 + k] = sc[k] * inv;
    }
    __syncthreads();

    // ---- AV: out[b, d, y, x0+p] = sum_{i,j} attn * Vs ----
    {
        const int p    = t & 15;
        const int dsub = t >> 4;                   // 0..7
        float* ob = og + (size_t)b * DIM * NN + (size_t)y * WW + x0;
        float w9[9];
        #pragma unroll
        for (int k = 0; k < 9; ++k) w9[k] = Attn[p * 9 + k];
        for (int dd = 0; dd < DIM / 8; ++dd) {
            int d = dsub + dd * 8;
            float o = 0.0f;
            #pragma unroll
            for (int i = 0; i < 3; ++i)
                #pragma unroll
                for (int j = 0; j < 3; ++j)
                    o += w9[i * 3 + j] * Vs[(i * DIM + d) * SPAN + p + 2 * j];
            ob[(size_t)d * NN + p] = o;
        }
    }
}

extern "C" void kernel_launch(void* const* d_in, const int* in_sizes, int n_in,
                              void* d_out, int out_size, void* d_ws, size_t ws_size,
                              hipStream_t stream) {
    (void)in_sizes; (void)n_in; (void)out_size; (void)d_ws; (void)ws_size;
    const float* q = (const float*)d_in[0];
    const float* k = (const float*)d_in[1];
    const float* v = (const float*)d_in[2];
    float* out = (float*)d_out;
    const int grid = BB * HH * (WW / TILE);   // 2048 workgroups
    dilated_attn_kernel<<<grid, 128, 0, stream>>>(q, k, v, out);
}